// EncoderLayer_75488345195383
// MI455X (gfx1250) — compile-verified
//
#include <hip/hip_runtime.h>
#include <math.h>

typedef __attribute__((ext_vector_type(2))) float v2f;
typedef __attribute__((ext_vector_type(8))) float v8f;

#define D_MODEL 512
#define D_FF    2048
#define NB      8
#define SEQ     1024
#define NROWS   (NB * SEQ)   /* 8192 */
#define HEADS   8
#define DKH     64

__device__ __forceinline__ v8f wmma_f32_k4(v2f a, v2f b, v8f c) {
  // D = A(16x4 f32) * B(4x16 f32) + C(16x16 f32)
  return __builtin_amdgcn_wmma_f32_16x16x4_f32(false, a, false, b, (short)0, c,
                                               false, false);
}

// ---------------------------------------------------------------------------
// Generic GEMM: C[M,N] = A[M,K] @ W[K,N] + bias[N] (+ res[M,N]) (+ ReLU)
// 128x64 C tile per 256-thread block: 8 wave32s in a 4x2 grid, each wave
// computes a 32x32 sub-tile = 2x2 WMMA tiles (4 v_wmma per k-step from
// 2 A + 2 B fragments -> 1.0 LDS fragment-loads per WMMA).
// M % 128 == 0, N % 64 == 0, K % 32 == 0.
// ---------------------------------------------------------------------------
__global__ __launch_bounds__(256) void gemm64(const float* __restrict__ A,
                                              const float* __restrict__ W,
                                              const float* __restrict__ bias,
                                              const float* __restrict__ res,
                                              float* __restrict__ C,
                                              int M, int N, int K, int relu) {
  const int ALD = 36;  // 32+4: gcd(36,64)=4 -> conflict-free, 16B-aligned
  const int BLD = 68;
  __shared__ __align__(16) float As[128 * ALD];
  __shared__ __align__(16) float Bs[32 * BLD];

  const int tid  = threadIdx.x;
  const int lane = tid & 31;
  const int w    = tid >> 5;
  const int m0   = blockIdx.y * 128;
  const int n0   = blockIdx.x * 64;
  const int wm   = w >> 1;   // wave row   (0..3) -> 32-row band
  const int wn   = w & 1;    // wave col   (0..1) -> 32-col band
  const int ml   = lane & 15;
  const int koff = (lane >> 4) << 1;  // 0 for lanes 0-15, 2 for lanes 16-31
  const int rsel = (lane >> 4) << 3;  // C-row offset per layout

  v8f acc[4] = {{0, 0, 0, 0, 0, 0, 0, 0},
                {0, 0, 0, 0, 0, 0, 0, 0},
                {0, 0, 0, 0, 0, 0, 0, 0},
                {0, 0, 0, 0, 0, 0, 0, 0}};

  for (int kb = 0; kb < K; kb += 32) {
#pragma unroll
    for (int r = 0; r < 4; ++r) {  // A tile: 128 x 32 (1024 float4)
      int idx = tid + 256 * r;
      int row = idx >> 3, c4 = (idx & 7) << 2;
      const float* gp = A + (size_t)(m0 + row) * K + kb + c4;
      *(float4*)(As + row * ALD + c4) = *(const float4*)gp;
      if (kb + 32 < K) __builtin_prefetch(gp + 32, 0, 0);
    }
#pragma unroll
    for (int r = 0; r < 2; ++r) {  // B tile: 32 x 64 (512 float4)
      int idx = tid + 256 * r;
      int row = idx >> 4, c4 = (idx & 15) << 2;
      const float* gp = W + (size_t)(kb + row) * N + n0 + c4;
      *(float4*)(Bs + row * BLD + c4) = *(const float4*)gp;
      if (kb + 32 < K) __builtin_prefetch(gp + (size_t)32 * N, 0, 0);
    }
    __syncthreads();

#pragma unroll
    for (int k = 0; k < 32; k += 4) {
      v2f a0, a1, b0, b1;
      int m = wm * 32 + ml;
      a0.x = As[m * ALD + k + koff];
      a0.y = As[m * ALD + k + koff + 1];
      a1.x = As[(m + 16) * ALD + k + koff];
      a1.y = As[(m + 16) * ALD + k + koff + 1];
      int n = wn * 32 + ml;
      b0.x = Bs[(k + koff) * BLD + n];
      b0.y = Bs[(k + koff + 1) * BLD + n];
      b1.x = Bs[(k + koff) * BLD + n + 16];
      b1.y = Bs[(k + koff + 1) * BLD + n + 16];
      acc[0] = wmma_f32_k4(a0, b0, acc[0]);
      acc[1] = wmma_f32_k4(a0, b1, acc[1]);
      acc[2] = wmma_f32_k4(a1, b0, acc[2]);
      acc[3] = wmma_f32_k4(a1, b1, acc[3]);
    }
    __syncthreads();
  }

#pragma unroll
  for (int tM = 0; tM < 2; ++tM)
#pragma unroll
    for (int tN = 0; tN < 2; ++tN) {
      int gcol = n0 + wn * 32 + tN * 16 + ml;
      float bv = bias[gcol];
#pragma unroll
      for (int i = 0; i < 8; ++i) {
        int grow = m0 + wm * 32 + tM * 16 + i + rsel;
        float v = acc[tM * 2 + tN][i] + bv;
        if (relu) v = fmaxf(v, 0.0f);
        if (res) v += res[(size_t)grow * N + gcol];
        C[(size_t)grow * N + gcol] = v;
      }
    }
}

// ---------------------------------------------------------------------------
// Flash-style attention, dk = 64. One block per (query-block of 64, b*h).
// Q/K/V/O are [B*S, 512] with head h occupying columns [h*64, h*64+64).
// ---------------------------------------------------------------------------
__global__ __launch_bounds__(256) void attn64(const float* __restrict__ Q,
                                              const float* __restrict__ K,
                                              const float* __restrict__ V,
                                              float* __restrict__ O) {
  const int LD = 68;
  __shared__ __align__(16) float Qs[64 * LD];
  __shared__ __align__(16) float KPs[64 * LD];  // K tile, then overlaid by P
  __shared__ __align__(16) float Vs[64 * LD];
  __shared__ float m_i[64], l_i[64], c_i[64];

  const int tid  = threadIdx.x;
  const int lane = tid & 31;
  const int w    = tid >> 5;
  const int qblk = blockIdx.x;
  const int bh   = blockIdx.y;
  const int b    = bh >> 3;
  const int h    = bh & 7;
  const size_t base = (size_t)b * SEQ * D_MODEL + (size_t)h * DKH;
  const int q0 = qblk * 64;

  const int tm   = w >> 1;
  const int tn0  = (w & 1) * 2;
  const int ml   = lane & 15;
  const int koff = (lane >> 4) << 1;
  const int rsel = (lane >> 4) << 3;

#pragma unroll
  for (int r = 0; r < 4; ++r) {  // Q tile 64 x 64 (1024 float4)
    int idx = tid + 256 * r;
    int row = idx >> 4, c4 = (idx & 15) << 2;
    *(float4*)(Qs + row * LD + c4) =
        *(const float4*)(Q + base + (size_t)(q0 + row) * D_MODEL + c4);
  }
  if (tid < 64) { m_i[tid] = -3.0e38f; l_i[tid] = 0.0f; }

  v8f oacc[2] = {{0, 0, 0, 0, 0, 0, 0, 0}, {0, 0, 0, 0, 0, 0, 0, 0}};
  __syncthreads();

  for (int j0 = 0; j0 < SEQ; j0 += 64) {
#pragma unroll
    for (int r = 0; r < 4; ++r) {  // K and V tiles (64 x 64 each)
      int idx = tid + 256 * r;
      int row = idx >> 4, c4 = (idx & 15) << 2;
      size_t g = base + (size_t)(j0 + row) * D_MODEL + c4;
      *(float4*)(KPs + row * LD + c4) = *(const float4*)(K + g);
      *(float4*)(Vs + row * LD + c4)  = *(const float4*)(V + g);
    }
    __syncthreads();

    // S = Q @ K^T   (B fragment = K-tile row-major read, contiguous)
    v8f sacc[2] = {{0, 0, 0, 0, 0, 0, 0, 0}, {0, 0, 0, 0, 0, 0, 0, 0}};
#pragma unroll
    for (int k = 0; k < DKH; k += 4) {
      v2f a, b0, b1;
      int m = tm * 16 + ml;
      a.x = Qs[m * LD + k + koff];
      a.y = Qs[m * LD + k + koff + 1];
      int nl = tn0 * 16 + ml;
      b0.x = KPs[nl * LD + k + koff];
      b0.y = KPs[nl * LD + k + koff + 1];
      b1.x = KPs[(nl + 16) * LD + k + koff];
      b1.y = KPs[(nl + 16) * LD + k + koff + 1];
      sacc[0] = wmma_f32_k4(a, b0, sacc[0]);
      sacc[1] = wmma_f32_k4(a, b1, sacc[1]);
    }
    __syncthreads();  // all waves done reading K tile before P overlays it

#pragma unroll
    for (int t = 0; t < 2; ++t)
#pragma unroll
      for (int i = 0; i < 8; ++i) {
        int row = tm * 16 + i + rsel, col = (tn0 + t) * 16 + ml;
        KPs[row * LD + col] = sacc[t][i] * 0.125f;  // 1/sqrt(64)
      }
    __syncthreads();

    if (tid < 64) {  // online softmax per query row
      int row = tid;
      float mx = m_i[row];
      for (int c = 0; c < 64; ++c) mx = fmaxf(mx, KPs[row * LD + c]);
      float corr = __expf(m_i[row] - mx);
      float sum = 0.0f;
      for (int c = 0; c < 64; ++c) {
        float p = __expf(KPs[row * LD + c] - mx);
        KPs[row * LD + c] = p;
        sum += p;
      }
      l_i[row] = corr * l_i[row] + sum;
      m_i[row] = mx;
      c_i[row] = corr;
    }
    __syncthreads();

#pragma unroll
    for (int t = 0; t < 2; ++t)
#pragma unroll
      for (int i = 0; i < 8; ++i)
        oacc[t][i] *= c_i[tm * 16 + i + rsel];

    // O += P @ V
#pragma unroll
    for (int k = 0; k < 64; k += 4) {
      v2f a, b0, b1;
      int m = tm * 16 + ml;
      a.x = KPs[m * LD + k + koff];
      a.y = KPs[m * LD + k + koff + 1];
      int nl = tn0 * 16 + ml;
      b0.x = Vs[(k + koff) * LD + nl];
      b0.y = Vs[(k + koff + 1) * LD + nl];
      b1.x = Vs[(k + koff) * LD + nl + 16];
      b1.y = Vs[(k + koff + 1) * LD + nl + 16];
      oacc[0] = wmma_f32_k4(a, b0, oacc[0]);
      oacc[1] = wmma_f32_k4(a, b1, oacc[1]);
    }
    __syncthreads();
  }

#pragma unroll
  for (int t = 0; t < 2; ++t)
#pragma unroll
    for (int i = 0; i < 8; ++i) {
      int row = tm * 16 + i + rsel, col = (tn0 + t) * 16 + ml;
      O[base + (size_t)(q0 + row) * D_MODEL + col] = oacc[t][i] / l_i[row];
    }
}

// ---------------------------------------------------------------------------
// LayerNorm over last dim (512). One block per row.
// ---------------------------------------------------------------------------
__global__ __launch_bounds__(256) void layernorm512(const float* __restrict__ in,
                                                    const float* __restrict__ g,
                                                    const float* __restrict__ bt,
                                                    float* __restrict__ out) {
  __shared__ float ss[256], sq[256];
  int row = blockIdx.x, tid = threadIdx.x;
  const float* p = in + (size_t)row * D_MODEL;
  float a = p[tid], b = p[tid + 256];
  ss[tid] = a + b;
  sq[tid] = a * a + b * b;
  __syncthreads();
  for (int s = 128; s > 0; s >>= 1) {
    if (tid < s) { ss[tid] += ss[tid + s]; sq[tid] += sq[tid + s]; }
    __syncthreads();
  }
  float mu  = ss[0] * (1.0f / D_MODEL);
  float var = sq[0] * (1.0f / D_MODEL) - mu * mu;
  float rs  = rsqrtf(var + 1e-5f);
  float* q = out + (size_t)row * D_MODEL;
  q[tid]       = (a - mu) * rs * g[tid]       + bt[tid];
  q[tid + 256] = (b - mu) * rs * g[tid + 256] + bt[tid + 256];
}

// ---------------------------------------------------------------------------
extern "C" void kernel_launch(void* const* d_in, const int* in_sizes, int n_in,
                              void* d_out, int out_size, void* d_ws,
                              size_t ws_size, hipStream_t stream) {
  (void)in_sizes; (void)n_in; (void)out_size; (void)ws_size;
  const float* x    = (const float*)d_in[0];
  // d_in[1] = pos_embed (unused by reference), d_in[2] = num_heads (fixed 8)
  const float* wq   = (const float*)d_in[3];
  const float* bq   = (const float*)d_in[4];
  const float* wk   = (const float*)d_in[5];
  const float* bk   = (const float*)d_in[6];
  const float* wv   = (const float*)d_in[7];
  const float* bv   = (const float*)d_in[8];
  const float* wo   = (const float*)d_in[9];
  const float* bo   = (const float*)d_in[10];
  const float* ln1g = (const float*)d_in[11];
  const float* ln1b = (const float*)d_in[12];
  const float* w1   = (const float*)d_in[13];
  const float* b1   = (const float*)d_in[14];
  const float* w2   = (const float*)d_in[15];
  const float* b2   = (const float*)d_in[16];
  const float* ln2g = (const float*)d_in[17];
  const float* ln2b = (const float*)d_in[18];
  float* out = (float*)d_out;
  float* ws  = (float*)d_ws;

  const size_t T = (size_t)NROWS * D_MODEL;  // 4M floats
  float* Qb  = ws;
  float* Kb  = ws + T;
  float* Vb  = ws + 2 * T;
  float* Ob  = ws + 3 * T;
  float* TMP = ws + 4 * T;
  float* X1  = ws + 5 * T;
  float* FF1 = ws;  // overlays Q/K/V/O (dead after O-projection); 4*T floats

  dim3 blk(256);
  dim3 g_dd(D_MODEL / 64, NROWS / 128);  // (8, 64)
  dim3 g_df(D_FF / 64, NROWS / 128);     // (32, 64)

  gemm64<<<g_dd, blk, 0, stream>>>(x, wq, bq, nullptr, Qb, NROWS, D_MODEL, D_MODEL, 0);
  gemm64<<<g_dd, blk, 0, stream>>>(x, wk, bk, nullptr, Kb, NROWS, D_MODEL, D_MODEL, 0);
  gemm64<<<g_dd, blk, 0, stream>>>(x, wv, bv, nullptr, Vb, NROWS, D_MODEL, D_MODEL, 0);

  attn64<<<dim3(SEQ / 64, NB * HEADS), blk, 0, stream>>>(Qb, Kb, Vb, Ob);

  gemm64<<<g_dd, blk, 0, stream>>>(Ob, wo, bo, x, TMP, NROWS, D_MODEL, D_MODEL, 0);
  layernorm512<<<NROWS, blk, 0, stream>>>(TMP, ln1g, ln1b, X1);

  gemm64<<<g_df, blk, 0, stream>>>(X1, w1, b1, nullptr, FF1, NROWS, D_FF, D_MODEL, 1);
  gemm64<<<g_dd, blk, 0, stream>>>(FF1, w2, b2, X1, TMP, NROWS, D_MODEL, D_FF, 0);
  layernorm512<<<NROWS, blk, 0, stream>>>(TMP, ln2g, ln2b, out);
}